// Features_4D_To_2D_19344532701687
// MI455X (gfx1250) — compile-verified
//
#include <hip/hip_runtime.h>
#include <stdint.h>

// im2col for in:[16,112,112,128] f32 -> out:[16, 55*55, 4*4*128] f32, K=4, S=2.
// Pure data movement (0 FLOPs). HBM floor: ~397MB NT stores + ~103MB loads
// (input kept resident in 192MB L2 via NT output stores) => ~21us @ 23.3 TB/s.
// CDNA5 path: Tensor Data Mover. One wave = one patch:
//   TENSOR_LOAD_TO_LDS  : 3-D tile (128 x 4 x 4 elems, strides 1/128/14336) -> 8KB LDS
//   TENSOR_STORE_FROM_LDS: 1-D tile (2048 elems) -> contiguous output row, TH=NT
// 2 TDM instructions per 16KB moved; descriptor setup is ~60 SALU (hidden).

typedef float        f32x4 __attribute__((ext_vector_type(4)));
typedef unsigned int u32x4 __attribute__((ext_vector_type(4)));
typedef int          i32x8 __attribute__((ext_vector_type(8)));
typedef int          i32x4 __attribute__((ext_vector_type(4)));

#define NPATCH   48400               // 16 * 55 * 55
#define WAVES_PB 8                   // 256 threads / wave32
#define NBLOCK   (NPATCH / WAVES_PB) // 6050 (exact)

#define CPOL_RT        0             // default temporal
#define CPOL_STORE_NT  1             // gfx12+ CPol: TH bits [2:0], TH_STORE_NT = 1

__global__ __launch_bounds__(256)
void im2col_tdm_kernel(const float* __restrict__ in, float* __restrict__ out) {
#if defined(__gfx1250__)
  __shared__ float buf[WAVES_PB][2048];   // 8KB per wave, 64KB per block

  const unsigned wv = (unsigned)__builtin_amdgcn_readfirstlane((int)(threadIdx.x >> 5));
  const unsigned p  = blockIdx.x * WAVES_PB + wv;      // patch id, uniform per wave
  const unsigned b  = p / 3025u;
  const unsigned r  = p - b * 3025u;
  const unsigned h  = r / 55u;                         // width-axis patch index
  const unsigned v  = r - h * 55u;                     // height-axis patch index

  // element offset of tile origin: ((b*112 + 2h)*112 + 2v)*128
  const uint64_t src = (uint64_t)(uintptr_t)in +
      4ull * (((uint64_t)(b * 112u + 2u * h) * 112u + 2u * v) * 128u);
  const uint64_t dst = (uint64_t)(uintptr_t)out + 8192ull * (uint64_t)p;

#if __has_builtin(__builtin_amdgcn_tensor_load_to_lds) && \
    __has_builtin(__builtin_amdgcn_tensor_store_from_lds) && \
    __has_builtin(__builtin_amdgcn_s_wait_tensorcnt)
  const uint32_t lds = (uint32_t)(uintptr_t)&buf[wv][0];

  // ---- D# for load: 3-D tile, data_size=4B ----
  // group0: [1:0]count=1 | lds_addr | global_addr[31:0] | global_addr[56:32] + type=2<<30
  u32x4 g0l = { 1u, lds, (uint32_t)src,
                ((uint32_t)(src >> 32) & 0x01FFFFFFu) | 0x80000000u };
  // group1 words:
  // w0: wg_mask=0 | data_size=2 (4B) <<16, no barrier/iterate/pad
  // w1: abar_addr[15:0]=0 | tensor_dim0[15:0]=128 <<16
  // w2: tensor_dim0[31:16]=0 | tensor_dim1[15:0]=112 <<16
  // w3: tensor_dim1[31:16]=0 | tile_dim0=128 <<16
  // w4: tile_dim1=4 | tile_dim2=4 <<16
  // w5: tensor_dim0_stride[31:0]=128     (k1 step)
  // w6: stride0[47:32]=0 | tensor_dim1_stride[15:0]=14336 <<16  (k0 step = 112*128)
  // w7: tensor_dim1_stride[47:16]=0
  i32x8 g1l = { (int)(2u << 16), (int)(128u << 16), (int)(112u << 16),
                (int)(128u << 16), (int)(4u | (4u << 16)), 128,
                (int)(14336u << 16), 0 };
  // group2: tensor_dim2=112 | tensor_dim3=0 | tensor_dim2_stride=112*112*128 | tile_dim3=0
  i32x4 g2l = { 112, 0, 1605632, 0 };
  i32x4 g3l = { 0, 0, 0, 0 };
  // Extra 8-DW group (encoding's VADDR4 slot; unused per ISA, pass zeros).
  i32x8 gz  = { 0, 0, 0, 0, 0, 0, 0, 0 };
  __builtin_amdgcn_tensor_load_to_lds(g0l, g1l, g2l, g3l, gz, CPOL_RT);

  // Per-wave tensor ops are in-order, but wait explicitly before the
  // dependent store reads LDS (cheap; many waves hide the latency).
  __builtin_amdgcn_s_wait_tensorcnt(0);

  // ---- D# for store: 1-D tile of 2048 contiguous elements, non-temporal ----
  u32x4 g0s = { 1u, lds, (uint32_t)dst,
                ((uint32_t)(dst >> 32) & 0x01FFFFFFu) | 0x80000000u };
  // tensor_dim0 = tile_dim0 = 2048, stride0 = 2048, dims 1..4 unused (tile_dim1=0)
  i32x8 g1s = { (int)(2u << 16), (int)(2048u << 16), 0,
                (int)(2048u << 16), 0, 2048, 0, 0 };
  i32x4 g2s = { 0, 0, 0, 0 };
  i32x4 g3s = { 0, 0, 0, 0 };
  __builtin_amdgcn_tensor_store_from_lds(g0s, g1s, g2s, g3s, gz, CPOL_STORE_NT);
  // S_ENDPGM implicitly waits for TENSORcnt==0.
#else
  // Fallback: same geometry, wave copies its 2048-float patch with float4s.
  (void)buf;
  const f32x4* __restrict__ sp = (const f32x4*)(uintptr_t)src;
  f32x4* __restrict__       dp = (f32x4*)(uintptr_t)dst;
  const unsigned lane = threadIdx.x & 31u;
  #pragma unroll
  for (unsigned q = lane; q < 512u; q += 32u) {   // 512 float4 per patch
    const unsigned d4 = q & 31u;
    const unsigned kk = q >> 5;                    // k0*4 + k1
    const unsigned k1 = kk & 3u, k0 = kk >> 2;
    __builtin_nontemporal_store(sp[(k0 * 112u + k1) * 32u + d4], dp + q);
  }
#endif
#else
  (void)in; (void)out;   // host pass / non-gfx1250: never executed
#endif
}

extern "C" void kernel_launch(void* const* d_in, const int* in_sizes, int n_in,
                              void* d_out, int out_size, void* d_ws, size_t ws_size,
                              hipStream_t stream) {
  (void)in_sizes; (void)n_in; (void)out_size; (void)d_ws; (void)ws_size;
  const float* in = (const float*)d_in[0];
  float* out = (float*)d_out;
  im2col_tdm_kernel<<<NBLOCK, 256, 0, stream>>>(in, out);
}